// SAGEConv_39565238731129
// MI455X (gfx1250) — compile-verified
//
#include <hip/hip_runtime.h>

#define N_NODES 50000
#define N_EDGES 800000
#define DIM 128

typedef __attribute__((ext_vector_type(2))) float v2f;
typedef __attribute__((ext_vector_type(8))) float v8f;

// ---------------------------------------------------------------------------
// Kernel 1: zero the workspace (agg[50000*128] + deg[50000], contiguous)
// ---------------------------------------------------------------------------
__global__ void __launch_bounds__(256) sage_zero(float* __restrict__ p, int n) {
  int i = blockIdx.x * 256 + threadIdx.x;
  if (i < n) p[i] = 0.0f;
}

// ---------------------------------------------------------------------------
// Kernel 2: per-edge scatter-add of RAW features (linearity rewrite):
//   agg[cols[e]] += feat[rows[e]];  deg[cols[e]] += 1
// One wave32 per edge; lane l handles 4 consecutive floats (32*4 = 128).
// feat/agg (25.6 MB each) are L2-resident on MI455X (192 MB L2), so this
// phase is L2-atomic bound, not HBM bound.
// ---------------------------------------------------------------------------
__global__ void __launch_bounds__(256) sage_scatter(const float* __restrict__ feat,
                                                    const int* __restrict__ rows,
                                                    const int* __restrict__ cols,
                                                    float* __restrict__ agg,
                                                    float* __restrict__ deg) {
  const int e    = (blockIdx.x * 256 + threadIdx.x) >> 5;
  const int lane = threadIdx.x & 31;
  if (e >= N_EDGES) return;

  const int r = rows[e];
  const int c = cols[e];
  if (lane == 0) atomicAdd(deg + c, 1.0f);

  const float4 v = *reinterpret_cast<const float4*>(feat + (size_t)r * DIM + lane * 4);
  float* dst = agg + (size_t)c * DIM + lane * 4;
  atomicAdd(dst + 0, v.x);
  atomicAdd(dst + 1, v.y);
  atomicAdd(dst + 2, v.z);
  atomicAdd(dst + 3, v.w);
}

// ---------------------------------------------------------------------------
// Kernel 3: fused output GEMM on the f32 WMMA pipe:
//   out[m][n] = sum_k feat[m][k]*w_self[n][k]
//            + (1/deg[m]) * sum_k agg[m][k]*w_neigh[n][k]
//            + b_self[n]
// Block = 256 threads = 8 wave32; block b covers node rows 16b..16b+15,
// wave w covers output cols 16w..16w+15. K=128 -> 32 steps of
// v_wmma_f32_16x16x4_f32 per weight matrix (64 WMMA per wave).
//
// ISA 7.12.2 fragment layouts (wave32):
//   A (16x4, MxK, 2 VGPR):  lanes 0-15 -> M=lane, K = {0,1};
//                           lanes 16-31 -> M=lane-16, K = {2,3}
//   B (4x16, KxN, 2 VGPR):  lanes 0-15 -> N=lane, K = {0,1} per VGPR;
//                           lanes 16-31 -> N=lane-16, K = {2,3}
//   C/D (16x16, 8 VGPR):    VGPR v: lanes 0-15 -> M=v, lanes 16-31 -> M=v+8
// ---------------------------------------------------------------------------
__global__ void __launch_bounds__(256) sage_gemm(const float* __restrict__ feat,
                                                 const float* __restrict__ w_neigh,
                                                 const float* __restrict__ w_self,
                                                 const float* __restrict__ b_self,
                                                 const float* __restrict__ agg,
                                                 const float* __restrict__ deg,
                                                 float* __restrict__ out) {
  const int lane = threadIdx.x & 31;
  const int wave = threadIdx.x >> 5;
  const int row0 = blockIdx.x * 16;   // node-row base of this block
  const int n0   = wave * 16;         // output-col base of this wave
  const int lo   = lane & 15;
  const int hi   = lane >> 4;         // 0: K={0,1}, 1: K={2,3}

  // 1/deg for the A-row this lane feeds (constant over the K loop)
  const float d = deg[row0 + lo];
  const float s = 1.0f / (d > 0.0f ? d : 1.0f);

  // accumulator initialized with broadcast bias: C[v][lane] = b[n0 + lo]
  const float bv = b_self[n0 + lo];
  v8f acc;
#pragma unroll
  for (int i = 0; i < 8; ++i) acc[i] = bv;

  const float* fA = feat    + (size_t)(row0 + lo) * DIM + hi * 2;  // A self
  const float* gA = agg     + (size_t)(row0 + lo) * DIM + hi * 2;  // A neigh
  const float* sB = w_self  + (size_t)(n0 + lo)   * DIM + hi * 2;  // B self
  const float* nB = w_neigh + (size_t)(n0 + lo)   * DIM + hi * 2;  // B neigh

#pragma unroll 4
  for (int k0 = 0; k0 < DIM; k0 += 4) {
    // self path: feat @ w_self^T
    v2f a0 = *reinterpret_cast<const v2f*>(fA + k0);
    v2f b0 = *reinterpret_cast<const v2f*>(sB + k0);
    acc = __builtin_amdgcn_wmma_f32_16x16x4_f32(false, a0, false, b0,
                                                (short)0, acc, false, false);
    // neighbor path: (agg * 1/deg) @ w_neigh^T  (row-scale folded into A)
    v2f a1 = *reinterpret_cast<const v2f*>(gA + k0);
    a1[0] *= s;
    a1[1] *= s;
    v2f b1 = *reinterpret_cast<const v2f*>(nB + k0);
    acc = __builtin_amdgcn_wmma_f32_16x16x4_f32(false, a1, false, b1,
                                                (short)0, acc, false, false);
  }

  // D layout: VGPR v -> row (row0 + v + 8*hi), col (n0 + lo)
  float* o = out + (size_t)(row0 + hi * 8) * DIM + n0 + lo;
#pragma unroll
  for (int v = 0; v < 8; ++v) o[(size_t)v * DIM] = acc[v];
}

// ---------------------------------------------------------------------------
extern "C" void kernel_launch(void* const* d_in, const int* in_sizes, int n_in,
                              void* d_out, int out_size, void* d_ws, size_t ws_size,
                              hipStream_t stream) {
  const float* feat    = (const float*)d_in[0];
  const float* w_neigh = (const float*)d_in[1];
  const float* w_self  = (const float*)d_in[2];
  const float* b_self  = (const float*)d_in[3];
  const int*   rows    = (const int*)d_in[4];
  const int*   cols    = (const int*)d_in[5];
  float* out = (float*)d_out;

  // workspace layout: agg [N_NODES*DIM] floats, then deg [N_NODES] floats
  float* agg = (float*)d_ws;
  float* deg = agg + (size_t)N_NODES * DIM;

  const int zero_n = N_NODES * DIM + N_NODES;
  sage_zero<<<(zero_n + 255) / 256, 256, 0, stream>>>(agg, zero_n);

  // one wave32 per edge -> 8 edges per 256-thread block
  const int scatter_blocks = (N_EDGES + 7) / 8;
  sage_scatter<<<scatter_blocks, 256, 0, stream>>>(feat, rows, cols, agg, deg);

  // 16 node-rows per block, 8 waves covering all 128 output cols
  sage_gemm<<<N_NODES / 16, 256, 0, stream>>>(feat, w_neigh, w_self, b_self,
                                              agg, deg, out);
}